// SVDHead_5729486372889
// MI455X (gfx1250) — compile-verified
//
#include <hip/hip_runtime.h>
#include <math.h>

typedef __attribute__((ext_vector_type(4))) float vf4;
typedef __attribute__((ext_vector_type(2))) float v2f;
typedef __attribute__((ext_vector_type(8))) float v8f;

#define BLOCK 256   // 8 waves of 32
#define NQ 22       // number of reduced scalars per batch

__device__ __forceinline__ vf4 ld_nt(const float* p) {
    return __builtin_nontemporal_load((const vf4*)p);
}

__device__ __forceinline__ void jacobi_rot(float A[3][3], float V[3][3], int p, int q) {
    float apq = A[p][q];
    if (fabsf(apq) < 1e-24f) return;
    float tau = (A[q][q] - A[p][p]) / (2.0f * apq);
    float t   = copysignf(1.0f, tau) / (fabsf(tau) + sqrtf(1.0f + tau * tau));
    float c   = 1.0f / sqrtf(1.0f + t * t);
    float s   = t * c;
    #pragma unroll
    for (int k = 0; k < 3; ++k) {   // column rotation
        float akp = A[k][p], akq = A[k][q];
        A[k][p] = c * akp - s * akq;
        A[k][q] = s * akp + c * akq;
    }
    #pragma unroll
    for (int k = 0; k < 3; ++k) {   // row rotation
        float apk = A[p][k], aqk = A[q][k];
        A[p][k] = c * apk - s * aqk;
        A[q][k] = s * apk + c * aqk;
    }
    #pragma unroll
    for (int k = 0; k < 3; ++k) {   // accumulate eigenvectors
        float vkp = V[k][p], vkq = V[k][q];
        V[k][p] = c * vkp - s * vkq;
        V[k][q] = s * vkp + c * vkq;
    }
}

__global__ __launch_bounds__(BLOCK)
void svdhead_kernel(const float* __restrict__ src,
                    const float* __restrict__ corr,
                    const float* __restrict__ wgt,
                    float* __restrict__ out,
                    int B, int N) {
    const int b   = blockIdx.x;
    const int tid = threadIdx.x;

    __shared__ float part[8][32];  // [wave][quantity], padded with zeros
    __shared__ float tot[32];      // block-level totals from WMMA

    const size_t base3 = (size_t)b * 3 * N;
    const float* sx = src  + base3;
    const float* sy = sx + N;
    const float* sz = sy + N;
    const float* cx = corr + base3;
    const float* cy = cx + N;
    const float* cz = cy + N;
    const float* ww = wgt + (size_t)b * N;

    // ---- streaming accumulation: 22 running sums per thread --------------
    float acc[NQ];
    #pragma unroll
    for (int q = 0; q < NQ; ++q) acc[q] = 0.0f;

    const int nvec = N >> 2;  // float4 groups
    for (int v = tid; v < nvec; v += BLOCK) {
        const int o = v << 2;
        vf4 SX = ld_nt(sx + o), SY = ld_nt(sy + o), SZ = ld_nt(sz + o);
        vf4 CX = ld_nt(cx + o), CY = ld_nt(cy + o), CZ = ld_nt(cz + o);
        vf4 W  = ld_nt(ww + o);
        #pragma unroll
        for (int k = 0; k < 4; ++k) {
            float sx_ = SX[k], sy_ = SY[k], sz_ = SZ[k];
            float cx_ = CX[k], cy_ = CY[k], cz_ = CZ[k];
            float w_  = W[k];
            float wx = w_ * sx_, wy = w_ * sy_, wz = w_ * sz_;
            acc[0]  += wx * cx_;  acc[1]  += wx * cy_;  acc[2]  += wx * cz_;
            acc[3]  += wy * cx_;  acc[4]  += wy * cy_;  acc[5]  += wy * cz_;
            acc[6]  += wz * cx_;  acc[7]  += wz * cy_;  acc[8]  += wz * cz_;
            acc[9]  += wx;        acc[10] += wy;        acc[11] += wz;
            acc[12] += w_ * cx_;  acc[13] += w_ * cy_;  acc[14] += w_ * cz_;
            acc[15] += sx_;       acc[16] += sy_;       acc[17] += sz_;
            acc[18] += cx_;       acc[19] += cy_;       acc[20] += cz_;
            acc[21] += w_;
        }
    }

    // ---- wave32 butterfly reduction --------------------------------------
    #pragma unroll
    for (int q = 0; q < NQ; ++q) {
        float v_ = acc[q];
        v_ += __shfl_xor(v_, 1);
        v_ += __shfl_xor(v_, 2);
        v_ += __shfl_xor(v_, 4);
        v_ += __shfl_xor(v_, 8);
        v_ += __shfl_xor(v_, 16);
        acc[q] = v_;
    }

    const int wv   = tid >> 5;
    const int lane = tid & 31;
    if (lane >= NQ) part[wv][lane] = 0.0f;          // zero-pad rows 22..31
    if (lane == 0) {
        #pragma unroll
        for (int q = 0; q < NQ; ++q) part[wv][q] = acc[q];
    }
    __syncthreads();

    // ---- cross-wave reduction via V_WMMA_F32_16X16X4_F32 -----------------
    // A(16x4) = partials[quantity][wave-chunk], B(4x16) = all-ones
    // => D[q][*] += sum over 4 waves. Two accumulating WMMAs cover 8 waves.
    // All-ones B makes the result independent of exact lane striping.
    if (tid < 32) {
        const int half = lane >> 4;     // 0: lanes 0-15, 1: lanes 16-31
        const int row  = lane & 15;
        v2f bones; bones[0] = 1.0f; bones[1] = 1.0f;
        v8f c1 = {}; v8f c2 = {};
        v2f a;
        // quantities 0..15, waves 0-3 then 4-7
        a[0] = part[half ? 2 : 0][row];
        a[1] = part[half ? 3 : 1][row];
        c1 = __builtin_amdgcn_wmma_f32_16x16x4_f32(false, a, false, bones,
                                                   (short)0, c1, false, false);
        a[0] = part[half ? 6 : 4][row];
        a[1] = part[half ? 7 : 5][row];
        c1 = __builtin_amdgcn_wmma_f32_16x16x4_f32(false, a, false, bones,
                                                   (short)0, c1, false, false);
        // quantities 16..31 (22..31 are zero padding)
        a[0] = part[half ? 2 : 0][16 + row];
        a[1] = part[half ? 3 : 1][16 + row];
        c2 = __builtin_amdgcn_wmma_f32_16x16x4_f32(false, a, false, bones,
                                                   (short)0, c2, false, false);
        a[0] = part[half ? 6 : 4][16 + row];
        a[1] = part[half ? 7 : 5][16 + row];
        c2 = __builtin_amdgcn_wmma_f32_16x16x4_f32(false, a, false, bones,
                                                   (short)0, c2, false, false);
        // D layout: lanes 0-15 hold M = vgpr (0..7); lanes 16-31 hold M = vgpr+8
        if (row == 0) {
            #pragma unroll
            for (int r = 0; r < 8; ++r) {
                tot[half * 8 + r]      = c1[r];
                tot[16 + half * 8 + r] = c2[r];
            }
        }
    }
    __syncthreads();

    // ---- finalize: assemble H, 3x3 SVD (Jacobi on H^T H), R & t ----------
    if (tid == 0) {
        float P[9], WS[3], WC[3], Sm[3], Cm[3];
        #pragma unroll
        for (int q = 0; q < 9; ++q) P[q] = tot[q];
        #pragma unroll
        for (int i = 0; i < 3; ++i) {
            WS[i] = tot[9 + i];
            WC[i] = tot[12 + i];
            Sm[i] = tot[15 + i];
            Cm[i] = tot[18 + i];
        }
        const float SW   = tot[21];
        const float invN = 1.0f / (float)N;
        float ms[3], mc[3];
        #pragma unroll
        for (int i = 0; i < 3; ++i) { ms[i] = Sm[i] * invN; mc[i] = Cm[i] * invN; }

        // H[i][j] = sum w*(src_i - ms_i)*(corr_j - mc_j), expanded
        float H[3][3];
        #pragma unroll
        for (int i = 0; i < 3; ++i)
            #pragma unroll
            for (int j = 0; j < 3; ++j)
                H[i][j] = P[i * 3 + j] - mc[j] * WS[i] - ms[i] * WC[j]
                        + ms[i] * mc[j] * SW;

        // S = H^T H (symmetric), Jacobi eigendecomposition -> V, eigenvalues
        float A[3][3];
        #pragma unroll
        for (int i = 0; i < 3; ++i)
            #pragma unroll
            for (int j = 0; j < 3; ++j)
                A[i][j] = H[0][i] * H[0][j] + H[1][i] * H[1][j] + H[2][i] * H[2][j];
        float V[3][3] = {{1,0,0},{0,1,0},{0,0,1}};
        for (int sweep = 0; sweep < 10; ++sweep) {
            jacobi_rot(A, V, 0, 1);
            jacobi_rot(A, V, 0, 2);
            jacobi_rot(A, V, 1, 2);
        }
        float e[3] = {A[0][0], A[1][1], A[2][2]};
        // sort eigenpairs descending (SVD convention: sigma_2 smallest)
        #pragma unroll
        for (int pass = 0; pass < 3; ++pass) {
            #pragma unroll
            for (int k = 0; k < 2; ++k) {
                if (e[k] < e[k + 1]) {
                    float te = e[k]; e[k] = e[k + 1]; e[k + 1] = te;
                    for (int i = 0; i < 3; ++i) {
                        float tv = V[i][k]; V[i][k] = V[i][k + 1]; V[i][k + 1] = tv;
                    }
                }
            }
        }
        // U columns: u_k = H v_k / ||H v_k||
        float U[3][3];
        #pragma unroll
        for (int k = 0; k < 3; ++k) {
            float u0 = H[0][0]*V[0][k] + H[0][1]*V[1][k] + H[0][2]*V[2][k];
            float u1 = H[1][0]*V[0][k] + H[1][1]*V[1][k] + H[1][2]*V[2][k];
            float u2 = H[2][0]*V[0][k] + H[2][1]*V[1][k] + H[2][2]*V[2][k];
            float nrm = sqrtf(u0*u0 + u1*u1 + u2*u2);
            if (nrm > 1e-20f) {
                float r = 1.0f / nrm;
                U[0][k] = u0 * r; U[1][k] = u1 * r; U[2][k] = u2 * r;
            } else if (k == 2) {  // degenerate: u2 = u0 x u1
                U[0][2] = U[1][0]*U[2][1] - U[2][0]*U[1][1];
                U[1][2] = U[2][0]*U[0][1] - U[0][0]*U[2][1];
                U[2][2] = U[0][0]*U[1][1] - U[1][0]*U[0][1];
            } else {
                U[0][k] = (k == 0); U[1][k] = (k == 1); U[2][k] = 0.0f;
            }
        }
        // d = sign(det(V U^T)) = sign(det H) since sigma >= 0
        float detH = H[0][0]*(H[1][1]*H[2][2] - H[1][2]*H[2][1])
                   - H[0][1]*(H[1][0]*H[2][2] - H[1][2]*H[2][0])
                   + H[0][2]*(H[1][0]*H[2][1] - H[1][1]*H[2][0]);
        float d = (detH < 0.0f) ? -1.0f : 1.0f;
        V[0][2] *= d; V[1][2] *= d; V[2][2] *= d;

        float R[3][3];
        #pragma unroll
        for (int i = 0; i < 3; ++i)
            #pragma unroll
            for (int j = 0; j < 3; ++j)
                R[i][j] = V[i][0]*U[j][0] + V[i][1]*U[j][1] + V[i][2]*U[j][2];

        float* Rout = out + (size_t)b * 9;
        #pragma unroll
        for (int i = 0; i < 3; ++i)
            #pragma unroll
            for (int j = 0; j < 3; ++j)
                Rout[i * 3 + j] = R[i][j];

        float* tout = out + (size_t)B * 9 + (size_t)b * 3;
        #pragma unroll
        for (int i = 0; i < 3; ++i)
            tout[i] = WC[i] - (R[i][0]*WS[0] + R[i][1]*WS[1] + R[i][2]*WS[2]);
    }
}

extern "C" void kernel_launch(void* const* d_in, const int* in_sizes, int n_in,
                              void* d_out, int out_size, void* d_ws, size_t ws_size,
                              hipStream_t stream) {
    const float* src  = (const float*)d_in[0];
    const float* corr = (const float*)d_in[1];
    const float* wgt  = (const float*)d_in[2];
    float* out = (float*)d_out;

    const int B = out_size / 12;            // 9 (R) + 3 (t) per batch
    const int N = in_sizes[2] / B;          // weights is (B, N)

    svdhead_kernel<<<dim3(B), dim3(BLOCK), 0, stream>>>(src, corr, wgt, out, B, N);
}